// RLTypologyMoE_53257594470429
// MI455X (gfx1250) — compile-verified
//
#include <hip/hip_runtime.h>

typedef __attribute__((ext_vector_type(2))) float v2f;
typedef __attribute__((ext_vector_type(8))) float v8f;

constexpr int Bn   = 64;
constexpr int Sn   = 512;
constexpr int Hn   = 768;
constexpr int TY   = 65;
constexpr int NE   = 4;
constexpr int NL   = 17;
constexpr int RH   = 256;   // router hidden
constexpr int LDA  = 772;   // LDS stride for seq tile (768+4 pad -> avoids 64-bank conflicts)
constexpr int LDT  = 68;    // LDS stride for typo tile (65 padded to 68, zero-filled)
constexpr int LDC2 = 80;    // 5 N-tiles of 16 covering E*L = 68 cols

__global__ __launch_bounds__(256)
void moe_fused(const float* __restrict__ seq,   // [B,S,H]
               const float* __restrict__ typo,  // [B,TY]
               const float* __restrict__ W1,    // [H+TY, RH]
               const float* __restrict__ b1,    // [RH]
               const float* __restrict__ W2,    // [RH, NE]
               const float* __restrict__ b2,    // [NE]
               const float* __restrict__ eW,    // [NE, H, NL]
               const float* __restrict__ eB,    // [NE, NL]
               float* __restrict__ out_final,   // [B*S, NL]
               float* __restrict__ out_alp,     // [B*S]
               float* __restrict__ out_act)     // [B*S]
{
    __shared__ float sA[16 * LDA];          // seq tile
    __shared__ float sT[16 * LDT];          // typo tile (zero padded cols 65..67)
    __shared__ float sPart[8 * 16 * NE];    // per-wave router-logit partials
    __shared__ float sC2[16 * LDC2];        // all-expert logits tile
    __shared__ int   sAct[16];

    const int tid = threadIdx.x;
    const int wv  = tid >> 5;
    const int ln  = tid & 31;                 // wave32 lane
    const int tileBase = blockIdx.x * 16;     // 16 tokens per block; S%16==0 so one sentence per tile row

    // ---- stage A tile into LDS ----
    for (int i = tid; i < 16 * (Hn / 4); i += 256) {      // 16 rows x 192 float4
        int r = i / (Hn / 4), c4 = i % (Hn / 4);
        float4 v = ((const float4*)(seq + (size_t)(tileBase + r) * Hn))[c4];
        *(float4*)(sA + r * LDA + c4 * 4) = v;
    }
    for (int i = tid; i < 16 * LDT; i += 256) {
        int r = i / LDT, c = i % LDT;
        int bb = (tileBase + r) >> 9;                     // token / S
        sT[r * LDT + c] = (c < TY) ? typo[bb * TY + c] : 0.0f;
    }
    for (int i = tid; i < 8 * 16 * NE; i += 256) sPart[i] = 0.0f;
    __syncthreads();

    const int tok    = ln & 15;
    const int rowsel = (ln >> 4) << 1;   // K sub-row select: 0 or 2
    const int crow   = (ln >> 4) << 3;   // C/D row offset: 0 or 8

    // ================= Phase 1: hdn = tanh(cat(seq,typo) @ W1 + b1), folded into router logits
    {
        const int n0a = wv * 32 + (ln & 15);
        const int n0b = n0a + 16;
        const float* pWa = W1 + (size_t)rowsel * RH + n0a;
        const float* pWb = W1 + (size_t)rowsel * RH + n0b;

        v8f acc0 = {}; v8f acc1 = {};
        // K = 0..767 : seq columns
        for (int k = 0; k < Hn; k += 4) {
            v2f a = *(const v2f*)(sA + tok * LDA + k + rowsel);
            v2f bA, bB;
            bA.x = pWa[(size_t)k * RH];       bA.y = pWa[(size_t)(k + 1) * RH];
            bB.x = pWb[(size_t)k * RH];       bB.y = pWb[(size_t)(k + 1) * RH];
            acc0 = __builtin_amdgcn_wmma_f32_16x16x4_f32(false, a, false, bA, (short)0, acc0, false, false);
            acc1 = __builtin_amdgcn_wmma_f32_16x16x4_f32(false, a, false, bB, (short)0, acc1, false, false);
        }
        // K = 768..831 : typo columns (all rows < 833, unguarded)
        for (int k = Hn; k < 832; k += 4) {
            v2f a = *(const v2f*)(sT + tok * LDT + (k - Hn) + rowsel);
            v2f bA, bB;
            bA.x = pWa[(size_t)k * RH];       bA.y = pWa[(size_t)(k + 1) * RH];
            bB.x = pWb[(size_t)k * RH];       bB.y = pWb[(size_t)(k + 1) * RH];
            acc0 = __builtin_amdgcn_wmma_f32_16x16x4_f32(false, a, false, bA, (short)0, acc0, false, false);
            acc1 = __builtin_amdgcn_wmma_f32_16x16x4_f32(false, a, false, bB, (short)0, acc1, false, false);
        }
        // epilogue chunk K = 832..835: only row 832 valid; LDS typo cols 65..67 are zero
        {
            v2f a = *(const v2f*)(sT + tok * LDT + (832 - Hn) + rowsel);
            v2f bA = {0.0f, 0.0f}, bB = {0.0f, 0.0f};
            if (rowsel == 0) {                     // lanes 0-15: row 832 in range
                bA.x = pWa[(size_t)832 * RH];
                bB.x = pWb[(size_t)832 * RH];
            }
            acc0 = __builtin_amdgcn_wmma_f32_16x16x4_f32(false, a, false, bA, (short)0, acc0, false, false);
            acc1 = __builtin_amdgcn_wmma_f32_16x16x4_f32(false, a, false, bB, (short)0, acc1, false, false);
        }

        // tanh + fold 256->4 projection into per-wave partial router logits
        float w2a[NE], w2b[NE];
        #pragma unroll
        for (int e = 0; e < NE; ++e) { w2a[e] = W2[n0a * NE + e]; w2b[e] = W2[n0b * NE + e]; }
        float bb1a = b1[n0a], bb1b = b1[n0b];
        #pragma unroll
        for (int r = 0; r < 8; ++r) {
            int row = r + crow;
            float h0 = tanhf(acc0[r] + bb1a);
            float h1 = tanhf(acc1[r] + bb1b);
            #pragma unroll
            for (int e = 0; e < NE; ++e)
                atomicAdd(&sPart[(wv * 16 + row) * NE + e], h0 * w2a[e] + h1 * w2b[e]);
        }
    }
    __syncthreads();

    // ================= Phase 2: log_softmax + argmax routing (one thread per token)
    if (tid < 16) {
        float x[NE];
        #pragma unroll
        for (int e = 0; e < NE; ++e) {
            float s = b2[e];
            #pragma unroll
            for (int w = 0; w < 8; ++w) s += sPart[(w * 16 + tid) * NE + e];
            x[e] = s;
        }
        float m = x[0]; int a = 0;
        #pragma unroll
        for (int e = 1; e < NE; ++e) if (x[e] > m) { m = x[e]; a = e; }   // first-max, matches jnp.argmax
        float sum = 0.0f;
        #pragma unroll
        for (int e = 0; e < NE; ++e) sum += expf(x[e] - m);
        float lse = m + logf(sum);
        int gt = tileBase + tid;
        out_alp[gt] = x[a] - lse;
        out_act[gt] = (float)a;
        sAct[tid] = a;
    }

    // ================= Phase 3: all-expert logits [16 x 68] via WMMA (waves 0..4)
    if (wv < 5) {
        int c = wv * 16 + (ln & 15);            // packed col = e*17 + l
        int e = 0, l = 0;
        if (c < NE * NL) { e = c / NL; l = c % NL; }   // pad cols read expert 0 harmlessly (never gathered)
        const float* pE = eW + ((size_t)e * Hn + rowsel) * NL + l;
        v8f acc = {};
        for (int k = 0; k < Hn; k += 4) {
            v2f a = *(const v2f*)(sA + tok * LDA + k + rowsel);
            v2f b;
            b.x = pE[(size_t)k * NL];
            b.y = pE[(size_t)(k + 1) * NL];
            acc = __builtin_amdgcn_wmma_f32_16x16x4_f32(false, a, false, b, (short)0, acc, false, false);
        }
        #pragma unroll
        for (int r = 0; r < 8; ++r) sC2[(r + crow) * LDC2 + c] = acc[r];
    }
    __syncthreads();

    // gather selected expert's 17 logits + bias
    for (int i = tid; i < 16 * NL; i += 256) {
        int t = i / NL, l = i % NL;
        int a = sAct[t];
        out_final[(size_t)(tileBase + t) * NL + l] = sC2[t * LDC2 + a * NL + l] + eB[a * NL + l];
    }
}

extern "C" void kernel_launch(void* const* d_in, const int* in_sizes, int n_in,
                              void* d_out, int out_size, void* d_ws, size_t ws_size,
                              hipStream_t stream) {
    const float* seq  = (const float*)d_in[0];
    const float* typo = (const float*)d_in[1];
    const float* W1   = (const float*)d_in[2];
    const float* b1   = (const float*)d_in[3];
    const float* W2   = (const float*)d_in[4];
    const float* b2   = (const float*)d_in[5];
    const float* eW   = (const float*)d_in[6];
    const float* eB   = (const float*)d_in[7];

    float* out       = (float*)d_out;
    float* out_final = out;                                  // [B*S*17]
    float* out_alp   = out + (size_t)Bn * Sn * NL;           // [B*S]
    float* out_act   = out_alp + (size_t)Bn * Sn;            // [B*S] (argmax as float)

    dim3 grid(Bn * Sn / 16), block(256);
    moe_fused<<<grid, block, 0, stream>>>(seq, typo, W1, b1, W2, b2, eW, eB,
                                          out_final, out_alp, out_act);
}